// MZ_opt_joint_map_all_86663850099033
// MI455X (gfx1250) — compile-verified
//
#include <hip/hip_runtime.h>

#define Mn   45
#define Bn   64
#define Tn   512
#define Hn   20
#define Gn   80
#define FC   32
#define MHD  256
#define OUTD 22
#define TOK  (Bn * Tn)

#define BQ   16            // batch rows per LSTM block
#define NBG  (Bn / BQ)     // 4 batch slices -> 45*4 = 180 blocks

typedef __attribute__((ext_vector_type(2))) float v2f;
typedef __attribute__((ext_vector_type(8))) float v8f;

__device__ __forceinline__ v8f wmma4(v2f a, v2f b, v8f c) {
  // V_WMMA_F32_16X16X4_F32 : D = A(16x4) * B(4x16) + C, full f32 precision
  return __builtin_amdgcn_wmma_f32_16x16x4_f32(false, a, false, b, (short)0, c,
                                               false, false);
}

// fast activations: single v_exp_f32 + v_rcp_f32 (TRANS ops, co-execute with XDL)
__device__ __forceinline__ float sigm(float x) {
  return __builtin_amdgcn_rcpf(1.f + __expf(-x));
}
__device__ __forceinline__ float tanh_fast(float x) {
  return 1.f - 2.f * __builtin_amdgcn_rcpf(1.f + __expf(2.f * x));
}

// ---------------------------------------------------------------------------
// Kernel 1: 45 independent 3-layer LSTMs + per-model readout head.
// One block = one (model, 16-batch slice); 5 waves, each owns one 16-gate
// N-tile. Weight (B) fragments live in VGPRs for the whole T loop; hidden/
// cell state + gate buffer live in LDS. Gate GEMMs: v_wmma_f32_16x16x4_f32.
// ---------------------------------------------------------------------------
__global__ __launch_bounds__(160) void lstm45_kernel(
    const float* __restrict__ Dis,
    const float* __restrict__ Wih0, const float* __restrict__ Whh0,
    const float* __restrict__ bih0, const float* __restrict__ bhh0,
    const float* __restrict__ Wih1, const float* __restrict__ Whh1,
    const float* __restrict__ bih1, const float* __restrict__ bhh1,
    const float* __restrict__ Wih2, const float* __restrict__ Whh2,
    const float* __restrict__ bih2, const float* __restrict__ bhh2,
    const float* __restrict__ Wfc1, const float* __restrict__ bfc1,
    const float* __restrict__ Wfc2, const float* __restrict__ bfc2,
    float* __restrict__ Dpred)
{
  __shared__ float xall[BQ][Tn];        // Dis slice, staged once (32 KB)
  __shared__ float hcat[BQ][3 * Hn];    // [b][ h0(20) | h1(20) | h2(20) ]
  __shared__ float ccat[BQ][3 * Hn];    // cell states, same layout
  __shared__ float gbuf[BQ][Gn];        // gate pre-activations
  __shared__ float zbuf[BQ][FC];        // head hidden
  __shared__ float wfc2s[FC];

  const int tid  = threadIdx.x;
  const int m    = blockIdx.y;          // model
  const int b0   = blockIdx.x * BQ;     // batch base
  const int lane = tid & 31;
  const int wv   = tid >> 5;            // 5 waves: one N-tile (16 gates) each
  const int lh   = lane & 15;
  const int sel  = lane >> 4;
  const int n    = wv * 16 + lh;        // gate column 0..79 (always valid)

  // ---- stage Dis slice into LDS (coalesced) ----
  for (int i = tid; i < BQ * Tn; i += 160) {
    int b = i / Tn, t = i % Tn;
    xall[b][t] = Dis[(m * Bn + b0 + b) * Tn + t];
  }
  // ---- zero h/c state ----
  for (int i = tid; i < BQ * 3 * Hn; i += 160) {
    (&hcat[0][0])[i] = 0.f;
    (&ccat[0][0])[i] = 0.f;
  }
  for (int f = tid; f < FC; f += 160) wfc2s[f] = Wfc2[m * FC + f];

  // ---- loop-invariant per-lane registers: biases + WMMA B-fragments ----
  const float bias0v = bih0[m * Gn + n] + bhh0[m * Gn + n];
  const float bias1v = bih1[m * Gn + n] + bhh1[m * Gn + n];
  const float bias2v = bih2[m * Gn + n] + bhh2[m * Gn + n];
  const float wih0v  = Wih0[m * Gn + n];
  const float bfc2v  = bfc2[m];

  const int wb = (m * Gn + n) * Hn;     // row base for 80x20 weight blocks
  v2f bw0[5], bw1[10], bw2[10];
  #pragma unroll
  for (int kk = 0; kk < 5; ++kk) {      // B(k,n) = W[n][k], k = 4kk+2sel(+1)
    const int k = 4 * kk + 2 * sel;
    bw0[kk][0] = Whh0[wb + k]; bw0[kk][1] = Whh0[wb + k + 1];
    bw1[kk][0] = Wih1[wb + k]; bw1[kk][1] = Wih1[wb + k + 1];
    bw2[kk][0] = Wih2[wb + k]; bw2[kk][1] = Wih2[wb + k + 1];
  }
  #pragma unroll
  for (int kk = 0; kk < 5; ++kk) {      // recurrent half (K rows 20..39)
    const int k = 4 * kk + 2 * sel;
    bw1[5 + kk][0] = Whh1[wb + k]; bw1[5 + kk][1] = Whh1[wb + k + 1];
    bw2[5 + kk][0] = Whh2[wb + k]; bw2[5 + kk][1] = Whh2[wb + k + 1];
  }
  v2f bwf[5];
  float bfc1v = 0.f;
  if (wv < 2) {                         // head tile: n < 32 only for waves 0,1
    const int fb = (m * FC + n) * Hn;
    #pragma unroll
    for (int kk = 0; kk < 5; ++kk) {
      const int k = 4 * kk + 2 * sel;
      bwf[kk][0] = Wfc1[fb + k]; bwf[kk][1] = Wfc1[fb + k + 1];
    }
    bfc1v = bfc1[m * FC + n];
  }

  // ---- precomputed (b,h) pairs for the LSTM-cell stage (trip count == 2) ---
  const int eb0 = tid / Hn,         eh0 = tid % Hn;          // element tid
  const int eb1 = (tid + 160) / Hn, eh1 = (tid + 160) % Hn;  // element tid+160

  __syncthreads();

  // one LSTM cell update (constant LDS offsets once b/h are registers)
#define CELL(l, bb, hh)                                                  \
  {                                                                      \
    float ig = sigm(gbuf[bb][hh]);                                       \
    float fg = sigm(gbuf[bb][Hn + (hh)]);                                \
    float gv = tanh_fast(gbuf[bb][2 * Hn + (hh)]);                       \
    float og = sigm(gbuf[bb][3 * Hn + (hh)]);                            \
    float c  = fg * ccat[bb][(l) * Hn + (hh)] + ig * gv;                 \
    ccat[bb][(l) * Hn + (hh)] = c;                                       \
    hcat[bb][(l) * Hn + (hh)] = og * tanh_fast(c);                       \
  }

  for (int t = 0; t < Tn; ++t) {
    // ---------- layer 0: bias + x*Wih0 + h0 @ Whh0^T  (K=20) ----------
    {
      v8f acc;
      #pragma unroll
      for (int r = 0; r < 8; ++r)
        acc[r] = bias0v + xall[r + 8 * sel][t] * wih0v;
      #pragma unroll
      for (int kk = 0; kk < 5; ++kk) {
        const int k = 4 * kk + 2 * sel;
        v2f a; a[0] = hcat[lh][k]; a[1] = hcat[lh][k + 1];
        acc = wmma4(a, bw0[kk], acc);
      }
      #pragma unroll
      for (int r = 0; r < 8; ++r) gbuf[r + 8 * sel][n] = acc[r];
    }
    __syncthreads();
    CELL(0, eb0, eh0)
    CELL(0, eb1, eh1)
    __syncthreads();

    // ---------- layer 1: [h0||h1] @ [Wih1^T;Whh1^T]  (K=40, contiguous) ----
    {
      v8f acc;
      #pragma unroll
      for (int r = 0; r < 8; ++r) acc[r] = bias1v;
      #pragma unroll
      for (int kk = 0; kk < 10; ++kk) {
        const int k = 4 * kk + 2 * sel;            // spans hcat cols 0..39
        v2f a; a[0] = hcat[lh][k]; a[1] = hcat[lh][k + 1];
        acc = wmma4(a, bw1[kk], acc);
      }
      #pragma unroll
      for (int r = 0; r < 8; ++r) gbuf[r + 8 * sel][n] = acc[r];
    }
    __syncthreads();
    CELL(1, eb0, eh0)
    CELL(1, eb1, eh1)
    __syncthreads();

    // ---------- layer 2: [h1||h2]  (hcat cols 20..59, contiguous) ----------
    {
      v8f acc;
      #pragma unroll
      for (int r = 0; r < 8; ++r) acc[r] = bias2v;
      #pragma unroll
      for (int kk = 0; kk < 10; ++kk) {
        const int k = Hn + 4 * kk + 2 * sel;
        v2f a; a[0] = hcat[lh][k]; a[1] = hcat[lh][k + 1];
        acc = wmma4(a, bw2[kk], acc);
      }
      #pragma unroll
      for (int r = 0; r < 8; ++r) gbuf[r + 8 * sel][n] = acc[r];
    }
    __syncthreads();
    CELL(2, eb0, eh0)
    CELL(2, eb1, eh1)
    __syncthreads();

    // ---------- head: z = relu(h2 @ Wfc1^T + bfc1)  (hcat cols 40..59) ----
    if (wv < 2) {                                  // wave-uniform branch
      v8f acc;
      #pragma unroll
      for (int r = 0; r < 8; ++r) acc[r] = bfc1v;
      #pragma unroll
      for (int kk = 0; kk < 5; ++kk) {
        const int k = 2 * Hn + 4 * kk + 2 * sel;
        v2f a; a[0] = hcat[lh][k]; a[1] = hcat[lh][k + 1];
        acc = wmma4(a, bwf[kk], acc);
      }
      #pragma unroll
      for (int r = 0; r < 8; ++r) zbuf[r + 8 * sel][n] = fmaxf(acc[r], 0.f);
    }
    __syncthreads();
    if (tid < BQ) {                                // y = z . wfc2 + bfc2
      float y = bfc2v;
      #pragma unroll
      for (int f = 0; f < FC; ++f) y += zbuf[tid][f] * wfc2s[f];
      Dpred[(m * Bn + b0 + tid) * Tn + t] = y;
    }
    // no trailing barrier needed: gbuf/zbuf hazards are covered by the
    // elem2->head and per-stage barriers of the next iteration
  }
#undef CELL
}

// ---------------------------------------------------------------------------
// Kernel 2: shared MLP 45 -> 256 -> 256 -> 256 -> 22 over 32768 tokens.
// 64-token tiles, activations ping-pong in padded LDS, weights from L2.
// ---------------------------------------------------------------------------
#define ASTR (MHD + 4)    // pad LDS stride: kills 16-way bank conflicts

__device__ __forceinline__ void hidden_layer(
    const float* __restrict__ W, const float* __restrict__ bias,
    const float (*IN)[ASTR], float (*OUT)[ASTR], int lane, int wv)
{
  const int lh = lane & 15, sel = lane >> 4;
  for (int tile = wv; tile < 64; tile += 8) {     // 4 M-tiles x 16 N-tiles
    const int mt = tile >> 4, nt = tile & 15;
    const int n  = nt * 16 + lh;
    const int mr = mt * 16 + lh;
    v8f acc;
    const float bv = bias[n];
    #pragma unroll
    for (int r = 0; r < 8; ++r) acc[r] = bv;
    #pragma unroll 4
    for (int kk = 0; kk < MHD / 4; ++kk) {
      const int k = 4 * kk + 2 * sel;
      v2f a, b;
      a[0] = IN[mr][k];      a[1] = IN[mr][k + 1];
      b[0] = W[n * MHD + k]; b[1] = W[n * MHD + k + 1];
      acc = wmma4(a, b, acc);
    }
    #pragma unroll
    for (int r = 0; r < 8; ++r)
      OUT[mt * 16 + r + 8 * sel][n] = fmaxf(acc[r], 0.f);
  }
}

__global__ __launch_bounds__(256) void mlp_kernel(
    const float* __restrict__ Dpred,
    const float* __restrict__ Wm0, const float* __restrict__ bm0,
    const float* __restrict__ Wm1, const float* __restrict__ bm1,
    const float* __restrict__ Wm2, const float* __restrict__ bm2,
    const float* __restrict__ Wm3, const float* __restrict__ bm3,
    float* __restrict__ Out)
{
  __shared__ float Abuf[64][ASTR];
  __shared__ float Bbuf[64][ASTR];

  const int tid  = threadIdx.x;
  const int lane = tid & 31, wv = tid >> 5;
  const int lh   = lane & 15, sel = lane >> 4;
  const int tok0 = blockIdx.x * 64;

  // gather features: feat[token][model] = D_pred[model][token]; pad 45 -> 48
  for (int i = tid; i < 64 * 48; i += 256) {
    int r = i / 48, f = i % 48;
    Abuf[r][f] = (f < Mn) ? Dpred[f * TOK + tok0 + r] : 0.f;
  }
  __syncthreads();

  // ---- layer 0: 45(K pad 48) -> 256, ReLU ----
  for (int tile = wv; tile < 64; tile += 8) {
    const int mt = tile >> 4, nt = tile & 15;
    const int n  = nt * 16 + lh;
    const int mr = mt * 16 + lh;
    v8f acc;
    const float bv = bm0[n];
    #pragma unroll
    for (int r = 0; r < 8; ++r) acc[r] = bv;
    #pragma unroll
    for (int kk = 0; kk < 12; ++kk) {
      const int k  = 4 * kk + 2 * sel;
      const int k0 = (k     < Mn) ? k     : (Mn - 1);  // clamp: A cols 45..47 are 0
      const int k1 = (k + 1 < Mn) ? k + 1 : (Mn - 1);
      v2f a, b;
      a[0] = Abuf[mr][k];      a[1] = Abuf[mr][k + 1];
      b[0] = Wm0[n * Mn + k0]; b[1] = Wm0[n * Mn + k1];
      acc = wmma4(a, b, acc);
    }
    #pragma unroll
    for (int r = 0; r < 8; ++r)
      Bbuf[mt * 16 + r + 8 * sel][n] = fmaxf(acc[r], 0.f);
  }
  __syncthreads();

  hidden_layer(Wm1, bm1, Bbuf, Abuf, lane, wv);   // layer 1
  __syncthreads();
  hidden_layer(Wm2, bm2, Abuf, Bbuf, lane, wv);   // layer 2
  __syncthreads();

  // ---- layer 3: 256 -> 22 (pad N to 32), write c_p_F [B,T,22] ----
  for (int tile = wv; tile < 8; tile += 8) {      // one C-tile per wave
    const int mt = tile >> 1, nt = tile & 1;
    const int n  = nt * 16 + lh;
    const int nc = (n < OUTD) ? n : (OUTD - 1);   // clamp pad lanes (store guarded)
    const int mr = mt * 16 + lh;
    v8f acc;
    const float bv = bm3[nc];
    #pragma unroll
    for (int r = 0; r < 8; ++r) acc[r] = bv;
    #pragma unroll 4
    for (int kk = 0; kk < MHD / 4; ++kk) {
      const int k = 4 * kk + 2 * sel;
      v2f a, b;
      a[0] = Bbuf[mr][k];       a[1] = Bbuf[mr][k + 1];
      b[0] = Wm3[nc * MHD + k]; b[1] = Wm3[nc * MHD + k + 1];
      acc = wmma4(a, b, acc);
    }
    if (n < OUTD) {
      #pragma unroll
      for (int r = 0; r < 8; ++r)
        Out[(tok0 + mt * 16 + r + 8 * sel) * OUTD + n] = acc[r];
    }
  }
}

// ---------------------------------------------------------------------------
extern "C" void kernel_launch(void* const* d_in, const int* in_sizes, int n_in,
                              void* d_out, int out_size, void* d_ws, size_t ws_size,
                              hipStream_t stream) {
  const float* Dis  = (const float*)d_in[0];
  const float* Wih0 = (const float*)d_in[1];
  const float* Whh0 = (const float*)d_in[2];
  const float* bih0 = (const float*)d_in[3];
  const float* bhh0 = (const float*)d_in[4];
  const float* Wih1 = (const float*)d_in[5];
  const float* Whh1 = (const float*)d_in[6];
  const float* bih1 = (const float*)d_in[7];
  const float* bhh1 = (const float*)d_in[8];
  const float* Wih2 = (const float*)d_in[9];
  const float* Whh2 = (const float*)d_in[10];
  const float* bih2 = (const float*)d_in[11];
  const float* bhh2 = (const float*)d_in[12];
  const float* Wfc1 = (const float*)d_in[13];
  const float* bfc1 = (const float*)d_in[14];
  const float* Wfc2 = (const float*)d_in[15];
  const float* bfc2 = (const float*)d_in[16];
  const float* Wm0  = (const float*)d_in[17];
  const float* bm0  = (const float*)d_in[18];
  const float* Wm1  = (const float*)d_in[19];
  const float* bm1  = (const float*)d_in[20];
  const float* Wm2  = (const float*)d_in[21];
  const float* bm2  = (const float*)d_in[22];
  const float* Wm3  = (const float*)d_in[23];
  const float* bm3  = (const float*)d_in[24];

  float* out   = (float*)d_out;
  float* Dpred = out;                    // output #1: [45,64,512]
  float* cpf   = out + Mn * Bn * Tn;     // output #2: [64,512,22]

  dim3 g1(NBG, Mn);                      // 4 batch slices x 45 models = 180 WGs
  lstm45_kernel<<<g1, 160, 0, stream>>>(
      Dis, Wih0, Whh0, bih0, bhh0, Wih1, Whh1, bih1, bhh1,
      Wih2, Whh2, bih2, bhh2, Wfc1, bfc1, Wfc2, bfc2, Dpred);

  mlp_kernel<<<TOK / 64, 256, 0, stream>>>(
      Dpred, Wm0, bm0, Wm1, bm1, Wm2, bm2, Wm3, bm3, cpf);
}